// BaseHead_25099788878655
// MI455X (gfx1250) — compile-verified
//
#include <hip/hip_runtime.h>

// Problem constants (fixed by the reference file)
#define SIZE     4096
#define START    1
#define STOP     256
#define DEPTH_C  2
#define BATCH    16
#define KROWS    253          // STOP - DEPTH - START
#define LMAX     4095         // SIZE - START
#define FLAT_LEN 1011840      // sum_{d=1..255} (4096-d)
#define OUT_LEN  1003651      // sum_{d=3..255} (4096-d)
#define NTHREADS 1024

typedef __attribute__((ext_vector_type(2))) float        v2f;
typedef __attribute__((ext_vector_type(8))) float        v8f;
typedef __attribute__((ext_vector_type(4))) unsigned int v4u;
typedef __attribute__((ext_vector_type(4))) int          v4i;
typedef __attribute__((ext_vector_type(8))) int          v8i;

__global__ __launch_bounds__(NTHREADS)
void diag_depth_kernel(const float* __restrict__ diag,
                       const float* __restrict__ left,
                       const float* __restrict__ right,
                       const float* __restrict__ cptr,
                       const int*   __restrict__ dptr,
                       float* __restrict__ out)
{
    // 16 batch rows x L f32, packed at stride L (<= 262,080 B). CDNA5: 320KB LDS/WGP.
    __shared__ float cur[BATCH * LMAX];
    __shared__ float scratch[LMAX];   // reused: 1024-wide reduction buffer, then row buffer
    __shared__ float m_sh;

    const int   tid   = threadIdx.x;
    const int   k     = blockIdx.x;          // diagonal row, 0..252
    const int   d     = k + 1;
    const int   L     = LMAX - k;            // length of this diagonal (>= 3843)
    const float c     = *cptr;
    const int   depth = *dptr;               // == 2

    const long off_in  = (long)k * LMAX             - (long)k * (k - 1) / 2;
    const long off_out = (long)k * (LMAX - DEPTH_C) - (long)k * (k - 1) / 2;

    // ---- load the whole 16 x L tile into LDS ----
#if __has_builtin(__builtin_amdgcn_tensor_load_to_lds)
    // One Tensor Data Mover descriptor: 2D tile, 16 rows of L f32,
    // global row stride = FLAT_LEN elements, LDS rows packed contiguously.
    if (tid < 32) {                          // single wave issues the DMA (EXEC ignored)
        unsigned long long ga = (unsigned long long)(uintptr_t)diag
                              + (unsigned long long)off_in * 4ull;
        unsigned int lds_addr = (unsigned int)(uintptr_t)&cur[0];

        // D# group 0: count=1 | lds_addr | global_addr[56:0] | type=2
        v4u g0 = { 1u,
                   lds_addr,
                   (unsigned int)ga,
                   ((unsigned int)(ga >> 32) & 0x01FFFFFFu) | 0x80000000u };

        // D# group 1: data_size=4B; tensor_dim0=L; tensor_dim1=16;
        // tile_dim0=L; tile_dim1=16; tile_dim2=0; tensor_dim0_stride=FLAT_LEN
        v8i g1 = { (int)(2u << 16),                              // data_size = 4B
                   (int)(((unsigned)L & 0xFFFFu) << 16),         // tensor_dim0[15:0]
                   (int)((16u << 16) | (((unsigned)L >> 16) & 0xFFFFu)), // dim1 lo | dim0 hi
                   (int)((unsigned)L << 16),                     // tile_dim0 (dim1 hi = 0)
                   16,                                           // tile_dim1 (tile_dim2 = 0)
                   (int)FLAT_LEN,                                // tensor_dim0_stride[31:0]
                   0, 0 };                                       // stride hi, dim1_stride

        v4i gz4 = { 0, 0, 0, 0 };
#if defined(__clang_major__) && __clang_major__ >= 23
        v8i gz8 = { 0, 0, 0, 0, 0, 0, 0, 0 };
        __builtin_amdgcn_tensor_load_to_lds(g0, g1, gz4, gz4, gz8, 0);
#else
        __builtin_amdgcn_tensor_load_to_lds(g0, g1, gz4, gz4, 0);
#endif
        __builtin_amdgcn_s_wait_tensorcnt(0);
    }
#else
    // Fallback: manual staged copy
    for (int b = 0; b < BATCH; ++b) {
        const float* src = diag + (long)b * FLAT_LEN + off_in;
        for (int j = tid; j < L; j += NTHREADS)
            cur[b * L + j] = src[j];
    }
#endif
    __syncthreads();

    for (int s = 0; s < depth; ++s) {
        const int Lv = L - s - 1;            // valid element count this step (>= 3840)

        // ---- pass 1: sum of log-values over all (b, j) ----
        float part = 0.0f;
        for (int b = 0; b < BATCH; ++b) {
            const float* R    = right + b * SIZE;
            const float* Lp   = left  + b * SIZE;
            const float* crow = cur   + b * L;
            for (int j = tid; j < Lv; j += NTHREADS) {
                int ir = j + d + s + 1; ir = (ir > SIZE - 1) ? (SIZE - 1) : ir;
                float nxt = 0.5f * c * (expf(crow[j]) * R[ir] + expf(crow[j + 1]) * Lp[j]);
                part += logf(nxt);
            }
        }
        scratch[tid] = part;
        __syncthreads();
        // tree-reduce 1024 -> 32
        for (int st = NTHREADS / 2; st >= 32; st >>= 1) {
            if (tid < st) scratch[tid] += scratch[tid + st];
            __syncthreads();
        }
        // final 32 -> 1 on wave 0 via V_WMMA_F32_16X16X4_F32:
        // A(16x4) holds the 32 partials, B(4x16)=ones => each D column sums all of A.
        if (tid < 32) {                      // wave-uniform branch: EXEC all-ones
            float v = scratch[tid];
            v2f A = { v,    0.0f };
            v2f B = { 1.0f, 1.0f };
            v8f C = {};
            v8f D = __builtin_amdgcn_wmma_f32_16x16x4_f32(
                        false, A, false, B, (short)0, C, false, false);
            float t = D[0] + D[1] + D[2] + D[3] + D[4] + D[5] + D[6] + D[7];
            t += __shfl_xor(t, 16, 32);      // combine M=0..7 and M=8..15 halves
            if (tid == 0) m_sh = t / ((float)BATCH * (float)Lv);
        }
        __syncthreads();
        const float m = m_sh;

        // ---- pass 2: recompute, normalize, write back per batch row ----
        // (stage through `scratch` so cur[j]/cur[j+1] reads finish before writes)
        for (int b = 0; b < BATCH; ++b) {
            const float* R    = right + b * SIZE;
            const float* Lp   = left  + b * SIZE;
            float*       crow = cur   + b * L;
            for (int j = tid; j < Lv; j += NTHREADS) {
                int ir = j + d + s + 1; ir = (ir > SIZE - 1) ? (SIZE - 1) : ir;
                float nxt = 0.5f * c * (expf(crow[j]) * R[ir] + expf(crow[j + 1]) * Lp[j]);
                scratch[j] = logf(nxt) - m;
            }
            __syncthreads();
            for (int j = tid; j < Lv; j += NTHREADS)
                crow[j] = scratch[j];
            __syncthreads();
        }
    }

    // ---- output: out[b][off_out + i] = cur[b][i], i < L - depth ----
    const int Lout = L - depth;
    for (int b = 0; b < BATCH; ++b) {
        float*       drow = out + (long)b * OUT_LEN + off_out;
        const float* crow = cur + b * L;
        for (int i = tid; i < Lout; i += NTHREADS)
            drow[i] = crow[i];
    }
}

extern "C" void kernel_launch(void* const* d_in, const int* in_sizes, int n_in,
                              void* d_out, int out_size, void* d_ws, size_t ws_size,
                              hipStream_t stream) {
    (void)in_sizes; (void)n_in; (void)d_ws; (void)ws_size; (void)out_size;
    const float* diag  = (const float*)d_in[0];
    const float* left  = (const float*)d_in[1];
    const float* right = (const float*)d_in[2];
    const float* cst   = (const float*)d_in[3];
    const int*   depth = (const int*)d_in[4];
    float*       out   = (float*)d_out;

    diag_depth_kernel<<<dim3(KROWS), dim3(NTHREADS), 0, stream>>>(
        diag, left, right, cst, depth, out);
}